// ConcatenateMeanMax_18640158064907
// MI455X (gfx1250) — compile-verified
//
#include <hip/hip_runtime.h>
#include <hip/hip_bf16.h>
#include <stdint.h>

#define DFT 64  // feature dim

// Exact pointee type the async-LDS builtins want: int __vector(4), in global
// (AS1, prints as __device__) and LDS (AS3) address spaces.
typedef int v4i __attribute__((ext_vector_type(4)));
typedef __attribute__((address_space(1))) v4i gv4i;
typedef __attribute__((address_space(3))) v4i lv4i;

// Monotone mapping float -> uint so unsigned max == float max.
__device__ __forceinline__ unsigned f2ord(float f) {
  unsigned u = __float_as_uint(f);
  return (u & 0x80000000u) ? ~u : (u | 0x80000000u);
}
__device__ __forceinline__ float ord2f(unsigned o) {
  return (o & 0x80000000u) ? __uint_as_float(o & 0x7fffffffu)
                           : __uint_as_float(~o);
}

// Hardware global_atomic_add_f32 (d_ws is coarse-grained device memory, so the
// "unsafe" (no fine-grained support) variant is exactly right and avoids a CAS loop).
__device__ __forceinline__ void atom_add_f32(float* p, float v) {
  unsafeAtomicAdd(p, v);
}

__device__ __forceinline__ void wait_async0() {
#if __has_builtin(__builtin_amdgcn_s_wait_asynccnt)
  __builtin_amdgcn_s_wait_asynccnt(0);
#else
  asm volatile("s_wait_asynccnt 0" ::: "memory");
#endif
}

// Passthrough copy (atom_ft, global_ft -> d_out) via CDNA5 async LDS engine.
// Each lane stages one float4 global->LDS->global; ASYNCcnt-tracked, no VGPR data.
__global__ void __launch_bounds__(256) copy_passthru(const float4* __restrict__ src,
                                                     float4* __restrict__ dst, int n) {
  __shared__ float4 stage[256];
  const int tid = threadIdx.x;
  const int stride = gridDim.x * blockDim.x;
  for (int i = blockIdx.x * blockDim.x + tid; i < n; i += stride) {
#if __has_builtin(__builtin_amdgcn_global_load_async_to_lds_b128) && \
    __has_builtin(__builtin_amdgcn_global_store_async_from_lds_b128)
    __builtin_amdgcn_global_load_async_to_lds_b128(
        (gv4i*)(src + i), (lv4i*)(stage + tid), 0, 0);
    wait_async0();
    __builtin_amdgcn_global_store_async_from_lds_b128(
        (gv4i*)(dst + i), (lv4i*)(stage + tid), 0, 0);
    wait_async0();
#else
    dst[i] = src[i];
#endif
  }
}

// One 32-lane slice per edge; each lane handles 2 consecutive features.
// sum: f32 atomic add; mx: u32 atomic max of ord-mapped floats; cnt: per-edge +1.
__global__ void __launch_bounds__(256) edge_accum(
    const float* __restrict__ src_ft, const int* __restrict__ src_idx,
    const int* __restrict__ dst_idx, float* __restrict__ sum,
    unsigned* __restrict__ mx, unsigned* __restrict__ cnt, int n_edges) {
  const int gid = blockIdx.x * blockDim.x + threadIdx.x;
  const int lane = gid & 31;
  const int stride_edges = (gridDim.x * blockDim.x) >> 5;
  for (int e = (gid >> 5); e < n_edges; e += stride_edges) {
    const int en = e + stride_edges;
    if (en < n_edges) {  // emits global_prefetch_b8 for next iteration's indices
      __builtin_prefetch(src_idx + en, 0, 0);
      __builtin_prefetch(dst_idx + en, 0, 0);
    }
    const int s = src_idx[e];
    const int d = dst_idx[e];
    const float2 v = *(const float2*)(src_ft + (size_t)s * DFT + lane * 2);
    const size_t b = (size_t)d * DFT + lane * 2;
    atom_add_f32(&sum[b], v.x);
    atom_add_f32(&sum[b + 1], v.y);
    atomicMax(&mx[b], f2ord(v.x));
    atomicMax(&mx[b + 1], f2ord(v.y));
    if (lane == 0) atomicAdd(&cnt[d], 1u);
  }
}

// Phase-A finalize: write bond_ft col [0,64), mean_a [64,128), max_a [128,192).
__global__ void __launch_bounds__(256) finalize_a(
    const float* __restrict__ bond_ft, const float* __restrict__ sum,
    const unsigned* __restrict__ mx, const unsigned* __restrict__ cnt,
    float* __restrict__ out_ft, int nb) {
  const int gid = blockIdx.x * blockDim.x + threadIdx.x;
  const int row = gid >> 5;
  if (row >= nb) return;
  const int lane = gid & 31;
  const unsigned c = cnt[row];
  const float inv = 1.0f / (float)(c ? c : 1u);
  const int j = lane * 2;
  const size_t ib = (size_t)row * DFT + j;
  const float2 s2 = *(const float2*)(sum + ib);
  const uint2 m2 = *(const uint2*)(mx + ib);
  const float2 bnd = *(const float2*)(bond_ft + ib);
  float2 mean; mean.x = s2.x * inv; mean.y = s2.y * inv;
  float2 mv;
  if (c) { mv.x = ord2f(m2.x); mv.y = ord2f(m2.y); } else { mv.x = 0.f; mv.y = 0.f; }
  float* o = out_ft + (size_t)row * (5 * DFT) + j;
  *(float2*)(o)           = bnd;
  *(float2*)(o + DFT)     = mean;
  *(float2*)(o + 2 * DFT) = mv;
}

// Phase-G finalize: write mean_g [192,256), max_g [256,320).
__global__ void __launch_bounds__(256) finalize_g(
    const float* __restrict__ sum, const unsigned* __restrict__ mx,
    const unsigned* __restrict__ cnt, float* __restrict__ out_ft, int nb) {
  const int gid = blockIdx.x * blockDim.x + threadIdx.x;
  const int row = gid >> 5;
  if (row >= nb) return;
  const int lane = gid & 31;
  const unsigned c = cnt[row];
  const float inv = 1.0f / (float)(c ? c : 1u);
  const int j = lane * 2;
  const size_t ib = (size_t)row * DFT + j;
  const float2 s2 = *(const float2*)(sum + ib);
  const uint2 m2 = *(const uint2*)(mx + ib);
  float2 mean; mean.x = s2.x * inv; mean.y = s2.y * inv;
  float2 mv;
  if (c) { mv.x = ord2f(m2.x); mv.y = ord2f(m2.y); } else { mv.x = 0.f; mv.y = 0.f; }
  float* o = out_ft + (size_t)row * (5 * DFT) + j;
  *(float2*)(o + 3 * DFT) = mean;
  *(float2*)(o + 4 * DFT) = mv;
}

extern "C" void kernel_launch(void* const* d_in, const int* in_sizes, int n_in,
                              void* d_out, int out_size, void* d_ws, size_t ws_size,
                              hipStream_t stream) {
  const float* atom_ft = (const float*)d_in[0];
  const float* bond_ft = (const float*)d_in[1];
  const float* glob_ft = (const float*)d_in[2];
  const int* a2b_src = (const int*)d_in[3];
  const int* a2b_dst = (const int*)d_in[4];
  const int* g2b_src = (const int*)d_in[5];
  const int* g2b_dst = (const int*)d_in[6];

  const int na = in_sizes[0] / DFT;
  const int nb = in_sizes[1] / DFT;
  const int ng = in_sizes[2] / DFT;
  const int ea = in_sizes[3];
  const int eg = in_sizes[5];

  // Output layout: atom_ft | ft (nb x 320) | global_ft
  float* out = (float*)d_out;
  float* out_atom = out;
  float* out_ft = out + (size_t)na * DFT;
  float* out_glob = out_ft + (size_t)nb * (5 * DFT);

  // Scratch (reused between phases A and G): sum | ordmax | cnt  (~258 MB)
  const size_t sum_elems = (size_t)nb * DFT;
  float* sum = (float*)d_ws;
  unsigned* mx = (unsigned*)(sum + sum_elems);
  unsigned* cnt = (unsigned*)(mx + sum_elems);
  const size_t ws_needed = (2 * sum_elems + (size_t)nb) * sizeof(float);

  // --- init A (all-zero bit patterns work for sum, ord-min, and count) ---
  (void)hipMemsetAsync(d_ws, 0, ws_needed, stream);

  // --- passthrough copies via async-LDS engine ---
  {
    const int n16 = (na * DFT) / 4;
    int blocks = (n16 + 255) / 256; if (blocks > 16384) blocks = 16384;
    copy_passthru<<<blocks, 256, 0, stream>>>((const float4*)atom_ft,
                                              (float4*)out_atom, n16);
  }
  {
    const int n16 = (ng * DFT) / 4;
    int blocks = (n16 + 255) / 256; if (blocks > 16384) blocks = 16384;
    copy_passthru<<<blocks, 256, 0, stream>>>((const float4*)glob_ft,
                                              (float4*)out_glob, n16);
  }

  // --- phase A: atom -> bond ---
  edge_accum<<<4096, 256, 0, stream>>>(atom_ft, a2b_src, a2b_dst, sum, mx, cnt, ea);
  {
    const int blocks = (int)(((long long)nb * 32 + 255) / 256);
    finalize_a<<<blocks, 256, 0, stream>>>(bond_ft, sum, mx, cnt, out_ft, nb);
  }

  // --- phase G: global -> bond (reuse scratch) ---
  (void)hipMemsetAsync(d_ws, 0, ws_needed, stream);
  edge_accum<<<4096, 256, 0, stream>>>(glob_ft, g2b_src, g2b_dst, sum, mx, cnt, eg);
  {
    const int blocks = (int)(((long long)nb * 32 + 255) / 256);
    finalize_g<<<blocks, 256, 0, stream>>>(sum, mx, cnt, out_ft, nb);
  }
}